// CorrTrajBlock_19988777795940
// MI455X (gfx1250) — compile-verified
//
#include <hip/hip_runtime.h>
#include <math.h>

typedef float v2f __attribute__((ext_vector_type(2)));
typedef float v8f __attribute__((ext_vector_type(8)));

// B=8, C=512, T=8, H=W=28, HW=784, S=T*HW=6272, R=64, K=4, Cq=128, P=T*R=512
#define BN  8
#define CN  512
#define TN  8
#define HWN 784
#define SN  6272
#define RN  64
#define KN  4
#define CQ  128
#define PN  512
#define CPAD 516   // C+2 padded to multiple of 4

static __device__ inline v8f zero8() {
  v8f z = {0.f,0.f,0.f,0.f,0.f,0.f,0.f,0.f};
  return z;
}

static __device__ inline v8f wmma4(v2f a, v2f b, v8f c) {
  // D = A(16x4) * B(4x16) + C, fp32
  return __builtin_amdgcn_wmma_f32_16x16x4_f32(false, a, false, b, (short)0, c, false, false);
}

// ---------------------------------------------------------------------------
// Stage 1: template_p = bn1(w_reduce @ x[:,:,0]) ; argmax over HW ; gather
// template_resample transposed into trA[b][r][c].
// ---------------------------------------------------------------------------
__global__ void k_template(const float* __restrict__ x, const float* __restrict__ wr,
                           const float* __restrict__ g, const float* __restrict__ be,
                           const float* __restrict__ mu, const float* __restrict__ va,
                           int* __restrict__ spt, float* __restrict__ trA) {
  int b = blockIdx.x >> 6;
  int r = blockIdx.x & 63;
  int tid = threadIdx.x; // 256
  float scale = g[r] * rsqrtf(va[r] + 1e-5f);
  float shift = be[r] - mu[r] * scale;

  float best = -3.4e38f;
  int bi = 0x7fffffff;
  for (int s = tid; s < HWN; s += 256) {
    float acc = 0.f;
    for (int c = 0; c < CN; ++c)
      acc += wr[r * CN + c] * x[(size_t)(b * CN + c) * SN + s];
    float v = acc * scale + shift;
    if (v > best || (v == best && s < bi)) { best = v; bi = s; }
  }
  __shared__ float sv[256];
  __shared__ int   si[256];
  sv[tid] = best; si[tid] = bi;
  __syncthreads();
  for (int o = 128; o > 0; o >>= 1) {
    if (tid < o) {
      if (sv[tid + o] > sv[tid] || (sv[tid + o] == sv[tid] && si[tid + o] < si[tid])) {
        sv[tid] = sv[tid + o]; si[tid] = si[tid + o];
      }
    }
    __syncthreads();
  }
  int ind = si[0];
  if (tid == 0) spt[blockIdx.x] = ind;
  for (int c = tid; c < CN; c += 256)
    trA[(size_t)(b * RN + r) * CN + c] = x[(size_t)(b * CN + c) * SN + ind];
}

// ---------------------------------------------------------------------------
// Stage 2: affinity[b][r][s] = sum_c trA[b][r][c] * x_flat[b][c][s]  (WMMA)
// ---------------------------------------------------------------------------
__global__ void k_affinity(const float* __restrict__ x, const float* __restrict__ trA,
                           float* __restrict__ aff) {
  int b = blockIdx.z, mt = blockIdx.y, nt = blockIdx.x;
  int lane = threadIdx.x & 31, half = lane >> 4, l = lane & 15;
  const float* A = trA + (size_t)b * RN * CN;   // 64 x 512
  const float* Bm = x  + (size_t)b * CN * SN;   // 512 x 6272
  float* Cm = aff + (size_t)b * RN * SN;
  int m0 = mt * 16, n0 = nt * 16;
  v8f acc = zero8();
  for (int k = 0; k < CN; k += 4) {
    int kk = k + half * 2;
    v2f a; a.x = A[(m0 + l) * CN + kk]; a.y = A[(m0 + l) * CN + kk + 1];
    v2f bb; bb.x = Bm[(size_t)kk * SN + n0 + l]; bb.y = Bm[(size_t)(kk + 1) * SN + n0 + l];
    acc = wmma4(a, bb, acc);
  }
#pragma unroll
  for (int v = 0; v < 8; ++v)
    Cm[(size_t)(m0 + v + half * 8) * SN + n0 + l] = acc[v];
}

// ---------------------------------------------------------------------------
// Stage 3: top-4 per (b,r,t) over HW (4 masked argmax passes)
// ---------------------------------------------------------------------------
__global__ void k_topk(const float* __restrict__ aff, int* __restrict__ topk) {
  int row = blockIdx.x;            // (b*64+r)*8+t
  int br = row >> 3, t = row & 7;
  const float* p = aff + (size_t)br * SN + t * HWN;
  int tid = threadIdx.x;
  __shared__ float sv[256];
  __shared__ int   si[256];
  __shared__ int   chosen[4];
  for (int j = 0; j < 4; ++j) {
    float best = -3.4e38f;
    int bi = 0x7fffffff;
    for (int s = tid; s < HWN; s += 256) {
      bool skip = false;
      for (int jj = 0; jj < j; ++jj) if (chosen[jj] == s) skip = true;
      if (skip) continue;
      float v = p[s];
      if (v > best || (v == best && s < bi)) { best = v; bi = s; }
    }
    sv[tid] = best; si[tid] = bi;
    __syncthreads();
    for (int o = 128; o > 0; o >>= 1) {
      if (tid < o) {
        if (sv[tid + o] > sv[tid] || (sv[tid + o] == sv[tid] && si[tid + o] < si[tid])) {
          sv[tid] = sv[tid + o]; si[tid] = si[tid + o];
        }
      }
      __syncthreads();
    }
    if (tid == 0) { chosen[j] = si[0]; topk[row * 4 + j] = si[0]; }
    __syncthreads();
  }
}

// ---------------------------------------------------------------------------
// Stage 4a: gather trajectories + coords into fuse[(b*4+k)][cpad=516][t*64+r]
// ---------------------------------------------------------------------------
__global__ void k_gather_fuse(const float* __restrict__ x, const int* __restrict__ topk,
                              float* __restrict__ fuse) {
  int t = blockIdx.x, k = blockIdx.y, b = blockIdx.z;
  int tid = threadIdx.x;
  int r = tid & 63, c4 = tid >> 6;
  int ind = topk[(((b * RN + r) * TN) + t) * 4 + k];
  int bk = b * KN + k;
  size_t obase = (size_t)bk * CPAD * PN;
  for (int cb = 0; cb < 128; ++cb) {
    int c = cb * 4 + c4;
    fuse[obase + (size_t)c * PN + t * RN + r] =
        x[((size_t)(b * CN + c) * TN + t) * HWN + ind];
  }
  if (c4 == 0) {
    fuse[obase + (size_t)512 * PN + t * RN + r] = (float)(ind / 28) * (1.0f / 28.0f);
    fuse[obase + (size_t)513 * PN + t * RN + r] = (float)(ind % 28) * (1.0f / 28.0f);
    fuse[obase + (size_t)514 * PN + t * RN + r] = 0.f;
    fuse[obase + (size_t)515 * PN + t * RN + r] = 0.f;
  }
}

// Pad w_proj (128 x 514) -> (128 x 516) with zeros
__global__ void k_pad_wproj(const float* __restrict__ wp, float* __restrict__ wpp) {
  int i = blockIdx.x * 256 + threadIdx.x;
  if (i >= CQ * CPAD) return;
  int q = i / CPAD, c = i % CPAD;
  wpp[i] = (c < CN + 2) ? wp[q * (CN + 2) + c] : 0.f;
}

// points[b][c][p] = mean_k traj  (from fuse rows c<512)
__global__ void k_points(const float* __restrict__ fuse, float* __restrict__ pts) {
  int i = blockIdx.x * 256 + threadIdx.x;
  if (i >= BN * CN * PN) return;
  int b = i / (CN * PN);
  int rem = i % (CN * PN);
  int c = rem / PN, p = rem % PN;
  float s = 0.f;
  for (int k = 0; k < KN; ++k)
    s += fuse[((size_t)(b * KN + k) * CPAD + c) * PN + p];
  pts[i] = s * 0.25f;
}

// ---------------------------------------------------------------------------
// Stage 4b: out2[bk][q][p] = sum_c wpp[q][c] * fuse[bk][c][p]   (WMMA, K=516)
// ---------------------------------------------------------------------------
__global__ void k_gemm2(const float* __restrict__ wpp, const float* __restrict__ fuse,
                        float* __restrict__ out2) {
  int bk = blockIdx.z, mt = blockIdx.y, nt = blockIdx.x;
  int lane = threadIdx.x & 31, half = lane >> 4, l = lane & 15;
  const float* Bm = fuse + (size_t)bk * CPAD * PN;
  float* Cm = out2 + (size_t)bk * CQ * PN;
  int m0 = mt * 16, n0 = nt * 16;
  v8f acc = zero8();
  for (int k = 0; k < CPAD; k += 4) {
    int kk = k + half * 2;
    v2f a; a.x = wpp[(m0 + l) * CPAD + kk]; a.y = wpp[(m0 + l) * CPAD + kk + 1];
    v2f bb; bb.x = Bm[(size_t)kk * PN + n0 + l]; bb.y = Bm[(size_t)(kk + 1) * PN + n0 + l];
    acc = wmma4(a, bb, acc);
  }
#pragma unroll
  for (int v = 0; v < 8; ++v)
    Cm[(size_t)(m0 + v + half * 8) * PN + n0 + l] = acc[v];
}

// bn2 then max over k -> fq[b][q][p]
__global__ void k_maxbn2(const float* __restrict__ out2,
                         const float* __restrict__ g, const float* __restrict__ be,
                         const float* __restrict__ mu, const float* __restrict__ va,
                         float* __restrict__ fq) {
  int i = blockIdx.x * 256 + threadIdx.x;
  if (i >= BN * CQ * PN) return;
  int b = i / (CQ * PN);
  int rem = i % (CQ * PN);
  int q = rem / PN, p = rem % PN;
  float scale = g[q] * rsqrtf(va[q] + 1e-5f);
  float shift = be[q] - mu[q] * scale;
  float m = -3.4e38f;
  for (int k = 0; k < KN; ++k) {
    float v = out2[((size_t)(b * KN + k) * CQ + q) * PN + p] * scale + shift;
    m = fmaxf(m, v);
  }
  fq[i] = m;
}

// ---------------------------------------------------------------------------
// Stage 5: temporal conv (3-tap over t, pad 1) + bn3 + relu, store TRANSPOSED
// tcT[b][p][o]  (WMMA; p-tiles never cross t since 64 % 16 == 0)
// ---------------------------------------------------------------------------
__global__ void k_conv(const float* __restrict__ wt, const float* __restrict__ fq,
                       const float* __restrict__ g, const float* __restrict__ be,
                       const float* __restrict__ mu, const float* __restrict__ va,
                       float* __restrict__ tcT) {
  int b = blockIdx.z, mt = blockIdx.y, nt = blockIdx.x;
  int lane = threadIdx.x & 31, half = lane >> 4, l = lane & 15;
  int m0 = mt * 16, n0 = nt * 16;
  int t = n0 >> 6;
  v8f acc = zero8();
  for (int dt = 0; dt < 3; ++dt) {
    int ts = t + dt - 1;
    if (ts < 0 || ts >= TN) continue;
    int coloff = (dt - 1) * RN;
    for (int k = 0; k < CQ; k += 4) {
      int kk = k + half * 2;
      v2f a;
      a.x = wt[((size_t)(m0 + l) * CQ + kk) * 3 + dt];
      a.y = wt[((size_t)(m0 + l) * CQ + kk + 1) * 3 + dt];
      v2f bb;
      bb.x = fq[((size_t)b * CQ + kk) * PN + n0 + coloff + l];
      bb.y = fq[((size_t)b * CQ + kk + 1) * PN + n0 + coloff + l];
      acc = wmma4(a, bb, acc);
    }
  }
#pragma unroll
  for (int v = 0; v < 8; ++v) {
    int o = m0 + v + half * 8;
    float scale = g[o] * rsqrtf(va[o] + 1e-5f);
    float shift = be[o] - mu[o] * scale;
    float val = fmaxf(acc[v] * scale + shift, 0.f);
    tcT[((size_t)b * PN + n0 + l) * CN + o] = val;   // transposed store
  }
}

// ---------------------------------------------------------------------------
// Stage 6 (fused, flagship): per (b, 32-col tile of S):
//   S1 = points^T(512x512) * Xtile(512x32)   -> LDS
//   softmax over the 512 rows of S1 per column
//   prop = tc(512x512) * S1                  (A read from tcT, coalesced)
//   out = x + prop
// Dynamic LDS: Xs(512*32) + Ss(512*32) floats = 128 KB -> 2 WGs per WGP.
// ---------------------------------------------------------------------------
__global__ void k_fused(const float* __restrict__ x, const float* __restrict__ pts,
                        const float* __restrict__ tcT, float* __restrict__ out) {
  extern __shared__ float smem[];
  float* Xs = smem;              // [512][32]
  float* Ss = smem + 512 * 32;   // [512][32]
  __shared__ float redmax[8][32];
  __shared__ float redsum[8][32];
  __shared__ float colmax[32];
  __shared__ float colsum[32];

  int b = blockIdx.y;
  int s0 = blockIdx.x * 32;
  int tid = threadIdx.x;           // 256 = 8 waves
  int wave = tid >> 5;
  int lane = tid & 31, half = lane >> 4, l = lane & 15;

  // load X tile (coalesced: 32 consecutive columns per row)
  {
    int n = tid & 31, c0 = tid >> 5;
    for (int c = c0; c < CN; c += 8)
      Xs[c * 32 + n] = x[(size_t)(b * CN + c) * SN + s0 + n];
  }
  __syncthreads();

  // GEMM1: S = points^T * X   (A[m=p][k=c] = pts[b][c][p], lane-coalesced)
  const float* P = pts + (size_t)b * CN * PN;
  int m0base = wave * 64;
  for (int nt = 0; nt < 2; ++nt) {
    int n0 = nt * 16;
    v8f acc[4] = {zero8(), zero8(), zero8(), zero8()};
    for (int k = 0; k < CN; k += 4) {
      int kk = k + half * 2;
      v2f bb; bb.x = Xs[kk * 32 + n0 + l]; bb.y = Xs[(kk + 1) * 32 + n0 + l];
#pragma unroll
      for (int mi = 0; mi < 4; ++mi) {
        int m0 = m0base + mi * 16;
        v2f a; a.x = P[(size_t)kk * PN + m0 + l]; a.y = P[(size_t)(kk + 1) * PN + m0 + l];
        acc[mi] = wmma4(a, bb, acc[mi]);
      }
    }
#pragma unroll
    for (int mi = 0; mi < 4; ++mi) {
      int m0 = m0base + mi * 16;
#pragma unroll
      for (int v = 0; v < 8; ++v)
        Ss[(m0 + v + half * 8) * 32 + n0 + l] = acc[mi][v];
    }
  }
  __syncthreads();

  // softmax over the 512 rows, per column (8 threads per column)
  {
    int col = tid & 31, q = tid >> 5;
    float m = -3.4e38f;
    for (int p = q * 64; p < q * 64 + 64; ++p) m = fmaxf(m, Ss[p * 32 + col]);
    redmax[q][col] = m;
    __syncthreads();
    if (tid < 32) {
      float mm = redmax[0][tid];
#pragma unroll
      for (int qq = 1; qq < 8; ++qq) mm = fmaxf(mm, redmax[qq][tid]);
      colmax[tid] = mm;
    }
    __syncthreads();
    float mm = colmax[col];
    float sum = 0.f;
    for (int p = q * 64; p < q * 64 + 64; ++p) {
      float e = __expf(Ss[p * 32 + col] - mm);
      Ss[p * 32 + col] = e;
      sum += e;
    }
    redsum[q][col] = sum;
    __syncthreads();
    if (tid < 32) {
      float ss = 0.f;
#pragma unroll
      for (int qq = 0; qq < 8; ++qq) ss += redsum[qq][tid];
      colsum[tid] = ss;
    }
    __syncthreads();
    float inv = 1.0f / colsum[col];
    for (int p = q * 64; p < q * 64 + 64; ++p) Ss[p * 32 + col] *= inv;
  }
  __syncthreads();

  // GEMM2: prop = tc * S  (A[m=o][k=p] = tcT[b][p][o], lane-coalesced)
  const float* TT = tcT + (size_t)b * PN * CN;
  for (int nt = 0; nt < 2; ++nt) {
    int n0 = nt * 16;
    v8f acc[4] = {zero8(), zero8(), zero8(), zero8()};
    for (int k = 0; k < PN; k += 4) {
      int kk = k + half * 2;
      v2f bb; bb.x = Ss[kk * 32 + n0 + l]; bb.y = Ss[(kk + 1) * 32 + n0 + l];
#pragma unroll
      for (int mi = 0; mi < 4; ++mi) {
        int m0 = m0base + mi * 16;
        v2f a; a.x = TT[(size_t)kk * CN + m0 + l]; a.y = TT[(size_t)(kk + 1) * CN + m0 + l];
        acc[mi] = wmma4(a, bb, acc[mi]);
      }
    }
#pragma unroll
    for (int mi = 0; mi < 4; ++mi) {
      int m0 = m0base + mi * 16;
#pragma unroll
      for (int v = 0; v < 8; ++v) {
        int m = m0 + v + half * 8;
        size_t idx = (size_t)(b * CN + m) * SN + s0 + n0 + l;
        out[idx] = x[idx] + acc[mi][v];
      }
    }
  }
}

// ---------------------------------------------------------------------------
extern "C" void kernel_launch(void* const* d_in, const int* in_sizes, int n_in,
                              void* d_out, int out_size, void* d_ws, size_t ws_size,
                              hipStream_t stream) {
  const float* x   = (const float*)d_in[0];
  const float* wr  = (const float*)d_in[1];
  const float* g1  = (const float*)d_in[2];
  const float* be1 = (const float*)d_in[3];
  const float* mu1 = (const float*)d_in[4];
  const float* va1 = (const float*)d_in[5];
  const float* wp  = (const float*)d_in[6];
  const float* g2  = (const float*)d_in[7];
  const float* be2 = (const float*)d_in[8];
  const float* mu2 = (const float*)d_in[9];
  const float* va2 = (const float*)d_in[10];
  const float* wt  = (const float*)d_in[11];
  const float* g3  = (const float*)d_in[12];
  const float* be3 = (const float*)d_in[13];
  const float* mu3 = (const float*)d_in[14];
  const float* va3 = (const float*)d_in[15];
  float* out = (float*)d_out;
  (void)in_sizes; (void)n_in; (void)out_size; (void)ws_size;

  char* base = (char*)d_ws;
  size_t off = 0;
  auto take = [&](size_t bytes) -> char* {
    char* p = base + off;
    off += (bytes + 255) & ~(size_t)255;
    return p;
  };
  float* trA  = (float*)take((size_t)BN * RN * CN * 4);        // 1.0 MB
  int*   spt  = (int*)  take((size_t)BN * RN * 4);
  float* aff  = (float*)take((size_t)BN * RN * SN * 4);        // 12.8 MB
  int*   topk = (int*)  take((size_t)BN * RN * TN * 4 * 4);
  float* fuse = (float*)take((size_t)BN * KN * CPAD * PN * 4); // 33.8 MB
  float* wpp  = (float*)take((size_t)CQ * CPAD * 4);
  float* out2 = (float*)take((size_t)BN * KN * CQ * PN * 4);   // 8.4 MB
  float* fq   = (float*)take((size_t)BN * CQ * PN * 4);        // 2.1 MB
  float* pts  = (float*)take((size_t)BN * CN * PN * 4);        // 8.4 MB
  float* tcT  = (float*)take((size_t)BN * PN * CN * 4);        // 8.4 MB

  // 1. template projection + argmax + gather
  k_template<<<dim3(BN * RN), dim3(256), 0, stream>>>(x, wr, g1, be1, mu1, va1, spt, trA);
  // 2. affinity GEMM (WMMA)
  k_affinity<<<dim3(SN / 16, RN / 16, BN), dim3(32), 0, stream>>>(x, trA, aff);
  // 3. top-4
  k_topk<<<dim3(BN * RN * TN), dim3(256), 0, stream>>>(aff, topk);
  // 4a. gather + coords + padding
  k_gather_fuse<<<dim3(TN, KN, BN), dim3(256), 0, stream>>>(x, topk, fuse);
  k_pad_wproj<<<dim3((CQ * CPAD + 255) / 256), dim3(256), 0, stream>>>(wp, wpp);
  k_points<<<dim3((BN * CN * PN) / 256), dim3(256), 0, stream>>>(fuse, pts);
  // 4b. projection GEMM (WMMA) + bn2 + max over k
  k_gemm2<<<dim3(PN / 16, CQ / 16, BN * KN), dim3(32), 0, stream>>>(wpp, fuse, out2);
  k_maxbn2<<<dim3((BN * CQ * PN) / 256), dim3(256), 0, stream>>>(out2, g2, be2, mu2, va2, fq);
  // 5. temporal conv GEMM (WMMA) + bn3 + relu, transposed store
  k_conv<<<dim3(PN / 16, CN / 16, BN), dim3(32), 0, stream>>>(wt, fq, g3, be3, mu3, va3, tcT);
  // 6. fused proj-GEMM + softmax + prop-GEMM + residual add (WMMA x2)
  k_fused<<<dim3(SN / 32, BN), dim3(256), 2 * 512 * 32 * sizeof(float), stream>>>(x, pts, tcT, out);
}